// SelfSupervisedLearning_31662498906487
// MI455X (gfx1250) — compile-verified
//
#include <hip/hip_runtime.h>
#include <hip/hip_bf16.h>

// Problem constants from the reference
#define B_DIM 8
#define S_DIM 128
#define NUM_RECO 64              // S * 0.5
#define HW 65536                 // 256*256, C=1
#define CHUNKS_PER_SLICE 16
#define THREADS 256
#define F4_PER_CHUNK (HW / (4 * CHUNKS_PER_SLICE))   // 1024 float4 per chunk
#define F4_PER_THREAD (F4_PER_CHUNK / THREADS)       // 4
#define NBLOCKS (B_DIM * NUM_RECO * CHUNKS_PER_SLICE) // 8192
#define TOTAL_ELEMS 33554432.0f  // 8*64*1*256*256

typedef __attribute__((ext_vector_type(2))) float v2f;
typedef __attribute__((ext_vector_type(8))) float v8f;

// ---------------------------------------------------------------------------
// Kernel 1: per-batch hit mask -> first 64 non-hit slice indices (ascending).
// ---------------------------------------------------------------------------
__global__ void remaining_kernel(const int* __restrict__ reco,
                                 int* __restrict__ remaining) {
  int b = threadIdx.x;
  if (b >= B_DIM) return;
  unsigned long long lo = 0ull, hi = 0ull;
  for (int j = 0; j < NUM_RECO; ++j) {
    int s = reco[b * NUM_RECO + j] & (S_DIM - 1);
    if (s < 64) lo |= 1ull << s;
    else        hi |= 1ull << (s - 64);
  }
  int cnt = 0;
  for (int s = 0; s < S_DIM && cnt < NUM_RECO; ++s) {
    bool hit = (s < 64) ? ((lo >> s) & 1ull) : ((hi >> (s - 64)) & 1ull);
    if (!hit) remaining[b * NUM_RECO + cnt++] = s;
  }
}

// ---------------------------------------------------------------------------
// Kernel 2: streaming |x| sum over selected slices. One 16KB chunk per block.
// Wave-level reduction via v_wmma_f32_16x16x4_f32 (A = 64 lane accumulators,
// B = ones -> D rows are row-sums; lane L(<16) 8-reg sum + lane L+16 8-reg sum
// equals the wave total).
// ---------------------------------------------------------------------------
__global__ void __launch_bounds__(THREADS)
abs_sum_kernel(const float* __restrict__ slices,
               const int* __restrict__ remaining,
               float* __restrict__ partials) {
  const int blk   = blockIdx.x;
  const int chunk = blk & (CHUNKS_PER_SLICE - 1);
  const int pair  = blk >> 4;            // (b, r) pair index, 0..511
  const int b     = pair >> 6;
  const int r     = pair & (NUM_RECO - 1);
  const int s     = remaining[b * NUM_RECO + r];

  const float4* base =
      (const float4*)(slices + (size_t)(b * S_DIM + s) * HW) +
      (size_t)chunk * F4_PER_CHUNK;

  // Speculative prefetch of the next chunk of this slice (global_prefetch_b8).
  if (chunk + 1 < CHUNKS_PER_SLICE)
    __builtin_prefetch(base + F4_PER_CHUNK + threadIdx.x, 0, 0);

  float acc0 = 0.0f, acc1 = 0.0f;   // two accumulators = the lane's 2 A-VGPRs
#pragma unroll
  for (int i = 0; i < F4_PER_THREAD; ++i) {
    float4 v = base[threadIdx.x + i * THREADS];
    acc0 += __builtin_fabsf(v.x) + __builtin_fabsf(v.y);
    acc1 += __builtin_fabsf(v.z) + __builtin_fabsf(v.w);
  }

  // Wave reduction: D = A x ones. EXEC is all-1s here (no divergence above
  // within a block). A(16x4 f32) = {acc0, acc1} per lane across 32 lanes.
  v2f a;  a.x = acc0; a.y = acc1;
  v2f bo; bo.x = 1.0f; bo.y = 1.0f;
  v8f c = {};
  c = __builtin_amdgcn_wmma_f32_16x16x4_f32(
      /*neg_a=*/false, a, /*neg_b=*/false, bo,
      /*c_mod=*/(short)0, c, /*reuse_a=*/false, /*reuse_b=*/false);
  float lane_sum = c[0] + c[1] + c[2] + c[3] + c[4] + c[5] + c[6] + c[7];

  __shared__ float red[THREADS];
  red[threadIdx.x] = lane_sum;
  __syncthreads();
  if (threadIdx.x == 0) {
    float tot = 0.0f;
#pragma unroll
    for (int w = 0; w < THREADS / 32; ++w)
      tot += red[w * 32 + 0] + red[w * 32 + 16];  // two half-wave totals
    partials[blk] = tot;
  }
}

// ---------------------------------------------------------------------------
// Kernel 3: fixed-order reduction of block partials -> scalar mean.
// ---------------------------------------------------------------------------
__global__ void __launch_bounds__(THREADS)
final_kernel(const float* __restrict__ partials, float* __restrict__ out) {
  __shared__ float red[THREADS];
  float acc = 0.0f;
  for (int i = threadIdx.x; i < NBLOCKS; i += THREADS) acc += partials[i];
  red[threadIdx.x] = acc;
  __syncthreads();
  if (threadIdx.x == 0) {
    float tot = 0.0f;
    for (int i = 0; i < THREADS; ++i) tot += red[i];
    out[0] = tot / TOTAL_ELEMS;
  }
}

extern "C" void kernel_launch(void* const* d_in, const int* in_sizes, int n_in,
                              void* d_out, int out_size, void* d_ws, size_t ws_size,
                              hipStream_t stream) {
  const float* slices = (const float*)d_in[0];
  // d_in[1] = transforms: provably dead (reconstructed slices are exactly 0).
  const int* reco = (const int*)d_in[2];

  int*   remaining = (int*)d_ws;                                     // 512 ints
  float* partials  = (float*)((char*)d_ws + B_DIM * NUM_RECO * sizeof(int)); // 8192 f32

  remaining_kernel<<<1, 32, 0, stream>>>(reco, remaining);
  abs_sum_kernel<<<NBLOCKS, THREADS, 0, stream>>>(slices, remaining, partials);
  final_kernel<<<1, THREADS, 0, stream>>>(partials, (float*)d_out);
}